// TFPerformerAttention_82240033783808
// MI455X (gfx1250) — compile-verified
//
#include <hip/hip_runtime.h>
#include <hip/hip_bf16.h>
#include <math.h>
#include <stdint.h>

#define DEVINL __device__ __forceinline__

namespace {

constexpr int Bb  = 2;
constexpr int Ls  = 4096;
constexpr int DMs = 512;
constexpr int Hh  = 8;
constexpr int Dh  = 64;     // per-head dim
constexpr int Mm  = 256;    // random features
constexpr int BH  = Bb * Hh;      // 16
constexpr int R2  = BH * Ls;      // 65536 rows of head-split activations
constexpr int NC  = Ls / 64;      // 64 chunks of 64
constexpr int LDA = 68;           // LDS row stride (floats): 68*4=272B, 16B-aligned rows

typedef __attribute__((ext_vector_type(2))) float v2f;
typedef __attribute__((ext_vector_type(8))) float v8f;

// D = A(16x4 f32) * B(4x16 f32) + C(16x16 f32), wave32
DEVINL v8f wmma4(v2f a, v2f b, v8f c) {
  return __builtin_amdgcn_wmma_f32_16x16x4_f32(
      /*neg_a=*/false, a, /*neg_b=*/false, b,
      /*c_mod=*/(short)0, c, /*reuse_a=*/false, /*reuse_b=*/false);
}

// A-matrix 16x4 f32 layout: lanes 0-15 -> M=lane, K={kb,kb+1}; lanes 16-31 -> K={kb+2,kb+3}
DEVINL v2f fragA(const float* As, int lane, int rbase, int kb) {
  const float* p = As + (size_t)(rbase + (lane & 15)) * LDA + kb + ((lane >> 4) << 1);
  v2f a; a.x = p[0]; a.y = p[1]; return a;
}
// B-matrix 4x16 f32: lanes 0-15 -> N=lane, K={kb,kb+1}; lanes 16-31 -> K={kb+2,kb+3}
DEVINL v2f fragB(const float* Bs, int lane, int cbase, int kb) {
  int k = kb + ((lane >> 4) << 1);
  int c = cbase + (lane & 15);
  v2f b; b.x = Bs[(size_t)k * LDA + c]; b.y = Bs[(size_t)(k + 1) * LDA + c]; return b;
}

DEVINL float4 ldg4(const float* p) { return *reinterpret_cast<const float4*>(p); }
DEVINL void   sts4(float* p, float4 v) { *reinterpret_cast<float4*>(p) = v; }

// ---- CDNA5 async global->LDS datapath (ASYNCcnt) -------------------------
// Per-lane 16B copy, LDS written directly by the memory system (no VGPR hop).
DEVINL uint32_t lds_off(const void* p) {  // generic addr low 32 bits == LDS offset
  return (uint32_t)(size_t)p;
}
DEVINL void async_load_b128(uint32_t ldsaddr, const float* gptr) {
  asm volatile("global_load_async_to_lds_b128 %0, %1, off"
               :: "v"(ldsaddr), "v"((uint64_t)(size_t)gptr)
               : "memory");
}
DEVINL void wait_async0() {
  asm volatile("s_wait_asynccnt 0" ::: "memory");
}

DEVINL float waveMax(float v) {
  for (int m = 16; m > 0; m >>= 1) v = fmaxf(v, __shfl_xor(v, m, 32));
  return v;
}
DEVINL float waveSum(float v) {
  for (int m = 16; m > 0; m >>= 1) v += __shfl_xor(v, m, 32);
  return v;
}

// ---------------------------------------------------------------------------
// K1: fused Q/K/V projection.  Out[b][h][l][d] = (X[r,:] @ W + bias)[n] * scale
// Double-buffered async global->LDS: next K-tile streams in during WMMA.
// grid (DM/64=8, R/64=128, 3 for q/k/v), block 256
// ---------------------------------------------------------------------------
__global__ __launch_bounds__(256) void k_proj(
    const float* __restrict__ Xq, const float* __restrict__ Xk, const float* __restrict__ Xv,
    const float* __restrict__ Wq, const float* __restrict__ Wk, const float* __restrict__ Wv,
    const float* __restrict__ bq, const float* __restrict__ bk, const float* __restrict__ bv,
    float* __restrict__ Qh, float* __restrict__ Kh, float* __restrict__ Vh) {
  __shared__ float As[2][64 * LDA];
  __shared__ float Bs[2][64 * LDA];
  const float* X; const float* W; const float* bias; float* Out; float scale;
  if (blockIdx.z == 0)      { X = Xq; W = Wq; bias = bq; Out = Qh; scale = 0.3535533905932738f; }
  else if (blockIdx.z == 1) { X = Xk; W = Wk; bias = bk; Out = Kh; scale = 0.3535533905932738f; }
  else                      { X = Xv; W = Wv; bias = bv; Out = Vh; scale = 1.0f; }

  const int tid = threadIdx.x, lane = tid & 31, wave = tid >> 5;
  const int r0 = (wave >> 2) * 32, c0 = (wave & 3) * 16;
  const int bx = blockIdx.x, by = blockIdx.y;

  auto issue = [&](int kb0, int buf) {
    for (int i = 0; i < 4; ++i) {
      int idx4 = tid + i * 256;
      int row = idx4 >> 4, c4 = (idx4 & 15) * 4;
      async_load_b128(lds_off(&As[buf][row * LDA + c4]),
                      &X[(size_t)(by * 64 + row) * DMs + kb0 + c4]);
      async_load_b128(lds_off(&Bs[buf][row * LDA + c4]),
                      &W[(size_t)(kb0 + row) * DMs + bx * 64 + c4]);
    }
  };

  issue(0, 0);
  wait_async0();
  __syncthreads();

  v8f acc0 = {}; v8f acc1 = {};
  for (int kb0 = 0; kb0 < DMs; kb0 += 64) {
    const int buf = (kb0 >> 6) & 1;
    if (kb0 + 64 < DMs) issue(kb0 + 64, buf ^ 1);   // prefetch next tile async
    for (int kk = 0; kk < 64; kk += 4) {
      v2f b  = fragB(Bs[buf], lane, c0, kk);
      v2f a0 = fragA(As[buf], lane, r0, kk);
      v2f a1 = fragA(As[buf], lane, r0 + 16, kk);
      acc0 = wmma4(a0, b, acc0);
      acc1 = wmma4(a1, b, acc1);
    }
    wait_async0();
    __syncthreads();
  }
  const int n = bx * 64 + c0 + (lane & 15);
  const float bi = bias[n];
  const int h = bx;            // tile width 64 == head dim
  const int d = n & 63;
  for (int s = 0; s < 2; ++s) {
    v8f acc = s ? acc1 : acc0;
    for (int j = 0; j < 8; ++j) {
      int r = by * 64 + r0 + s * 16 + j + 8 * (lane >> 4);
      int b_ = r >> 12, l = r & (Ls - 1);
      Out[(((size_t)b_ * Hh + h) * Ls + l) * Dh + d] = (acc[j] + bi) * scale;
    }
  }
}

// ---------------------------------------------------------------------------
// K2: feature logits  P[r][m] = sum_d A[r][d] * rf[m][d]  (A = Qh or Kh, [R2,64])
// grid (M/64=4, R2/64=1024, 2), block 256
// ---------------------------------------------------------------------------
__global__ __launch_bounds__(256) void k_phi(
    const float* __restrict__ Qh, const float* __restrict__ Kh,
    const float* __restrict__ rf,
    float* __restrict__ PQ, float* __restrict__ PK) {
  __shared__ float As[64 * LDA];
  __shared__ float Bs[64 * LDA];
  const float* A = blockIdx.z ? Kh : Qh;
  float* Out = blockIdx.z ? PK : PQ;
  const int tid = threadIdx.x, lane = tid & 31, wave = tid >> 5;
  const int r0 = (wave >> 2) * 32, c0 = (wave & 3) * 16;
  const int bx = blockIdx.x, by = blockIdx.y;

  for (int i = 0; i < 4; ++i) {
    int idx4 = tid + i * 256;
    int row = idx4 >> 4, c4 = (idx4 & 15) * 4;
    sts4(&As[row * LDA + c4], ldg4(&A[(size_t)(by * 64 + row) * Dh + c4]));
    float4 v = ldg4(&rf[(size_t)(bx * 64 + row) * Dh + c4]);   // row=m_local, c4=d
    Bs[(c4 + 0) * LDA + row] = v.x;
    Bs[(c4 + 1) * LDA + row] = v.y;
    Bs[(c4 + 2) * LDA + row] = v.z;
    Bs[(c4 + 3) * LDA + row] = v.w;
  }
  __syncthreads();
  v8f acc0 = {}; v8f acc1 = {};
  for (int kk = 0; kk < 64; kk += 4) {
    v2f b  = fragB(Bs, lane, c0, kk);
    v2f a0 = fragA(As, lane, r0, kk);
    v2f a1 = fragA(As, lane, r0 + 16, kk);
    acc0 = wmma4(a0, b, acc0);
    acc1 = wmma4(a1, b, acc1);
  }
  const int m = bx * 64 + c0 + (lane & 15);
  for (int s = 0; s < 2; ++s) {
    v8f acc = s ? acc1 : acc0;
    for (int j = 0; j < 8; ++j) {
      int r = by * 64 + r0 + s * 16 + j + 8 * (lane >> 4);
      Out[(size_t)r * Mm + m] = acc[j];
    }
  }
}

// ---------------------------------------------------------------------------
// K3/K4: global max of PK (two-stage tree)
// ---------------------------------------------------------------------------
__global__ __launch_bounds__(256) void k_max1(const float* __restrict__ PK,
                                              float* __restrict__ bmax) {
  __shared__ float red[8];
  size_t base = (size_t)blockIdx.x * 4096 + threadIdx.x;
  float m = -3.4e38f;
  for (int i = 0; i < 16; ++i) m = fmaxf(m, PK[base + (size_t)i * 256]);
  m = waveMax(m);
  int lane = threadIdx.x & 31, wave = threadIdx.x >> 5;
  if (lane == 0) red[wave] = m;
  __syncthreads();
  if (wave == 0) {
    float v = (lane < 8) ? red[lane] : -3.4e38f;
    v = waveMax(v);
    if (lane == 0) bmax[blockIdx.x] = v;
  }
}
__global__ __launch_bounds__(256) void k_max2(const float* __restrict__ bmax,
                                              float* __restrict__ kstab) {
  __shared__ float red[8];
  float m = -3.4e38f;
  for (int i = 0; i < 16; ++i) m = fmaxf(m, bmax[threadIdx.x + i * 256]);
  m = waveMax(m);
  int lane = threadIdx.x & 31, wave = threadIdx.x >> 5;
  if (lane == 0) red[wave] = m;
  __syncthreads();
  if (wave == 0) {
    float v = (lane < 8) ? red[lane] : -3.4e38f;
    v = waveMax(v);
    if (lane == 0) kstab[0] = v;
  }
}

// ---------------------------------------------------------------------------
// K5: feature map  P := norm*(exp(h + P - stab) + eps)  in place, one wave/row
// h = -0.5*||x||^2 from Xh row; stab = row max (q) or global scalar (k)
// grid (R2/8), block 256
// ---------------------------------------------------------------------------
__global__ __launch_bounds__(256) void k_expmap(
    float* __restrict__ P, const float* __restrict__ Xh,
    const float* __restrict__ kstab, int useRowMax) {
  const int lane = threadIdx.x & 31, wave = threadIdx.x >> 5;
  const size_t row = (size_t)blockIdx.x * 8 + wave;
  float* prow = P + row * Mm;
  float vals[8];
  float mx = -3.4e38f;
  for (int i = 0; i < 8; ++i) {
    vals[i] = prow[lane + 32 * i];
    mx = fmaxf(mx, vals[i]);
  }
  float x0 = Xh[row * Dh + lane], x1 = Xh[row * Dh + lane + 32];
  float hrow = -0.5f * waveSum(x0 * x0 + x1 * x1);
  float stab = useRowMax ? waveMax(mx) : kstab[0];
  for (int i = 0; i < 8; ++i)
    prow[lane + 32 * i] = 0.0625f * (expf(hrow + vals[i] - stab) + 1e-4f);
}

// ---------------------------------------------------------------------------
// K6: per-chunk sums  G[bh][c][m][d] = sum_t k'[t][m] v[t][d],  ZS = sum_t k'[t][m]
// grid (4 m-tiles, NC, BH), block 256
// ---------------------------------------------------------------------------
__global__ __launch_bounds__(256) void k_chunksum(
    const float* __restrict__ KP, const float* __restrict__ Vh,
    float* __restrict__ G, float* __restrict__ ZS) {
  __shared__ float As[64 * LDA];  // [m_local][t]
  __shared__ float Bs[64 * LDA];  // [t][d]
  const int tid = threadIdx.x, lane = tid & 31, wave = tid >> 5;
  const int r0 = (wave >> 2) * 32, c0 = (wave & 3) * 16;
  const int bx = blockIdx.x, c = blockIdx.y, bh = blockIdx.z;
  const size_t kbase = (size_t)bh * Ls + c * 64;
  for (int i = 0; i < 4; ++i) {
    int idx4 = tid + i * 256;
    int t = idx4 >> 4, c4 = (idx4 & 15) * 4;
    float4 v = ldg4(&KP[(kbase + t) * Mm + bx * 64 + c4]);
    As[(c4 + 0) * LDA + t] = v.x;
    As[(c4 + 1) * LDA + t] = v.y;
    As[(c4 + 2) * LDA + t] = v.z;
    As[(c4 + 3) * LDA + t] = v.w;
    sts4(&Bs[t * LDA + c4], ldg4(&Vh[(kbase + t) * Dh + c4]));
  }
  __syncthreads();
  v8f acc0 = {}; v8f acc1 = {};
  for (int kk = 0; kk < 64; kk += 4) {
    v2f b  = fragB(Bs, lane, c0, kk);
    v2f a0 = fragA(As, lane, r0, kk);
    v2f a1 = fragA(As, lane, r0 + 16, kk);
    acc0 = wmma4(a0, b, acc0);
    acc1 = wmma4(a1, b, acc1);
  }
  if (tid < 64) {
    float s = 0.f;
    for (int t = 0; t < 64; ++t) s += As[tid * LDA + t];
    ZS[((size_t)bh * NC + c) * Mm + bx * 64 + tid] = s;
  }
  const int d = c0 + (lane & 15);
  for (int s = 0; s < 2; ++s) {
    v8f acc = s ? acc1 : acc0;
    for (int j = 0; j < 8; ++j) {
      int mg = bx * 64 + r0 + s * 16 + j + 8 * (lane >> 4);
      G[(((size_t)bh * NC + c) * Mm + mg) * Dh + d] = acc[j];
    }
  }
}

// ---------------------------------------------------------------------------
// K7: exclusive prefix over chunks, in place (parallel over the M*D state)
// grid (64, BH), block 256
// ---------------------------------------------------------------------------
__global__ __launch_bounds__(256) void k_prefix(float* __restrict__ G,
                                                float* __restrict__ ZS) {
  const int bh = blockIdx.y;
  const int flat = blockIdx.x * 256 + threadIdx.x;  // 0..16383 over (m,d)
  float cur = 0.f;
  for (int c = 0; c < NC; ++c) {
    size_t idx = ((size_t)bh * NC + c) * (Mm * Dh) + flat;
    float g = G[idx]; G[idx] = cur; cur += g;
  }
  if (blockIdx.x == 0) {
    float cz = 0.f;
    for (int c = 0; c < NC; ++c) {
      size_t idx = ((size_t)bh * NC + c) * Mm + threadIdx.x;
      float z = ZS[idx]; ZS[idx] = cz; cz += z;
    }
  }
}

// ---------------------------------------------------------------------------
// K8: per-chunk output:
//   num = q' @ S_prefix + tril(q' @ k'^T) @ v ;  den = q'.z_prefix + rowsum(tril A)
//   ATT = num / stabilized(den)
// grid (NC, BH), block 256
// ---------------------------------------------------------------------------
__global__ __launch_bounds__(256) void k_outchunk(
    const float* __restrict__ QP, const float* __restrict__ KP,
    const float* __restrict__ Vh, const float* __restrict__ G,
    const float* __restrict__ ZS, float* __restrict__ ATT) {
  __shared__ float As[64 * LDA];    // q' tile [t][k]
  __shared__ float Bs1[64 * LDA];   // S tile [k][d], later v tile [t'][d]
  __shared__ float Bs2[64 * LDA];   // k'^T tile [k=m][t']
  __shared__ float Amat[64 * LDA];  // masked intra scores [t][t']
  __shared__ float zp[256];
  __shared__ float denp[64];
  __shared__ float dens[64];
  const int tid = threadIdx.x, lane = tid & 31, wave = tid >> 5;
  const int r0 = (wave >> 2) * 32, c0 = (wave & 3) * 16;
  const int c = blockIdx.x, bh = blockIdx.y;
  const size_t rbase = (size_t)bh * Ls + c * 64;

  zp[tid] = ZS[((size_t)bh * NC + c) * Mm + tid];
  if (tid < 64) denp[tid] = 0.f;
  __syncthreads();

  v8f n0 = {}, n1 = {}, aA0 = {}, aA1 = {};
  for (int kb = 0; kb < 4; ++kb) {
    for (int i = 0; i < 4; ++i) {
      int idx4 = tid + i * 256;
      int row = idx4 >> 4, c4 = (idx4 & 15) * 4;
      sts4(&As[row * LDA + c4],  ldg4(&QP[(rbase + row) * Mm + kb * 64 + c4]));
      sts4(&Bs1[row * LDA + c4], ldg4(&G[(((size_t)bh * NC + c) * Mm + kb * 64 + row) * Dh + c4]));
      float4 v = ldg4(&KP[(rbase + row) * Mm + kb * 64 + c4]);  // row=t', c4=m_local
      Bs2[(c4 + 0) * LDA + row] = v.x;
      Bs2[(c4 + 1) * LDA + row] = v.y;
      Bs2[(c4 + 2) * LDA + row] = v.z;
      Bs2[(c4 + 3) * LDA + row] = v.w;
    }
    __syncthreads();
    if (tid < 64) {
      float s = 0.f;
      for (int kk = 0; kk < 64; ++kk) s += As[tid * LDA + kk] * zp[kb * 64 + kk];
      denp[tid] += s;
    }
    for (int kk = 0; kk < 64; kk += 4) {
      v2f a0 = fragA(As, lane, r0, kk);
      v2f a1 = fragA(As, lane, r0 + 16, kk);
      v2f b1 = fragB(Bs1, lane, c0, kk);
      v2f b2 = fragB(Bs2, lane, c0, kk);
      n0  = wmma4(a0, b1, n0);
      n1  = wmma4(a1, b1, n1);
      aA0 = wmma4(a0, b2, aA0);
      aA1 = wmma4(a1, b2, aA1);
    }
    __syncthreads();
  }
  // masked store of intra scores; reload Bs1 with v
  {
    const int tp = c0 + (lane & 15);
    for (int s = 0; s < 2; ++s) {
      v8f acc = s ? aA1 : aA0;
      for (int j = 0; j < 8; ++j) {
        int t = r0 + s * 16 + j + 8 * (lane >> 4);
        Amat[t * LDA + tp] = (tp <= t) ? acc[j] : 0.f;
      }
    }
    for (int i = 0; i < 4; ++i) {
      int idx4 = tid + i * 256;
      int row = idx4 >> 4, c4 = (idx4 & 15) * 4;
      sts4(&Bs1[row * LDA + c4], ldg4(&Vh[(rbase + row) * Dh + c4]));
    }
  }
  __syncthreads();
  if (tid < 64) {
    float rs = 0.f;
    for (int t2 = 0; t2 < 64; ++t2) rs += Amat[tid * LDA + t2];
    float den = denp[tid] + rs;
    if (fabsf(den) <= 1e-6f) den += 2e-6f;   // reference stabilizer
    dens[tid] = den;
  }
  __syncthreads();
  for (int kk = 0; kk < 64; kk += 4) {
    v2f a0 = fragA(Amat, lane, r0, kk);
    v2f a1 = fragA(Amat, lane, r0 + 16, kk);
    v2f b  = fragB(Bs1, lane, c0, kk);
    n0 = wmma4(a0, b, n0);
    n1 = wmma4(a1, b, n1);
  }
  const int d = c0 + (lane & 15);
  for (int s = 0; s < 2; ++s) {
    v8f acc = s ? n1 : n0;
    for (int j = 0; j < 8; ++j) {
      int t = r0 + s * 16 + j + 8 * (lane >> 4);
      ATT[(rbase + t) * Dh + d] = acc[j] / dens[t];
    }
  }
}

// ---------------------------------------------------------------------------
// K9: output projection  Y[r][n] = att_flat[r,:] @ Wo + bo   (gather from [B,H,L,D])
// Double-buffered async global->LDS like K1.
// grid (8, 128), block 256
// ---------------------------------------------------------------------------
__global__ __launch_bounds__(256) void k_outproj(
    const float* __restrict__ ATT, const float* __restrict__ Wo,
    const float* __restrict__ bo, float* __restrict__ Y) {
  __shared__ float As[2][64 * LDA];
  __shared__ float Bs[2][64 * LDA];
  const int tid = threadIdx.x, lane = tid & 31, wave = tid >> 5;
  const int r0 = (wave >> 2) * 32, c0 = (wave & 3) * 16;
  const int bx = blockIdx.x, by = blockIdx.y;
  const int b_ = (by * 64) >> 12;
  const int l0 = (by * 64) & (Ls - 1);

  auto issue = [&](int kb0, int buf) {
    const int h = kb0 >> 6;
    for (int i = 0; i < 4; ++i) {
      int idx4 = tid + i * 256;
      int row = idx4 >> 4, c4 = (idx4 & 15) * 4;
      async_load_b128(lds_off(&As[buf][row * LDA + c4]),
                      &ATT[(((size_t)b_ * Hh + h) * Ls + l0 + row) * Dh + c4]);
      async_load_b128(lds_off(&Bs[buf][row * LDA + c4]),
                      &Wo[(size_t)(kb0 + row) * DMs + bx * 64 + c4]);
    }
  };

  issue(0, 0);
  wait_async0();
  __syncthreads();

  v8f acc0 = {}; v8f acc1 = {};
  for (int kb0 = 0; kb0 < DMs; kb0 += 64) {
    const int buf = (kb0 >> 6) & 1;
    if (kb0 + 64 < DMs) issue(kb0 + 64, buf ^ 1);
    for (int kk = 0; kk < 64; kk += 4) {
      v2f b  = fragB(Bs[buf], lane, c0, kk);
      v2f a0 = fragA(As[buf], lane, r0, kk);
      v2f a1 = fragA(As[buf], lane, r0 + 16, kk);
      acc0 = wmma4(a0, b, acc0);
      acc1 = wmma4(a1, b, acc1);
    }
    wait_async0();
    __syncthreads();
  }
  const int n = bx * 64 + c0 + (lane & 15);
  const float bi = bo[n];
  for (int s = 0; s < 2; ++s) {
    v8f acc = s ? acc1 : acc0;
    for (int j = 0; j < 8; ++j) {
      int r = by * 64 + r0 + s * 16 + j + 8 * (lane >> 4);
      Y[(size_t)r * DMs + n] = acc[j] + bi;
    }
  }
}

}  // namespace

extern "C" void kernel_launch(void* const* d_in, const int* in_sizes, int n_in,
                              void* d_out, int out_size, void* d_ws, size_t ws_size,
                              hipStream_t stream) {
  (void)in_sizes; (void)n_in; (void)out_size; (void)ws_size;
  const float* query = (const float*)d_in[0];
  const float* key_  = (const float*)d_in[1];
  const float* value = (const float*)d_in[2];
  const float* Wq = (const float*)d_in[3];
  const float* bq = (const float*)d_in[4];
  const float* Wk = (const float*)d_in[5];
  const float* bk = (const float*)d_in[6];
  const float* Wv = (const float*)d_in[7];
  const float* bv = (const float*)d_in[8];
  const float* Wo = (const float*)d_in[9];
  const float* bo = (const float*)d_in[10];
  const float* rf = (const float*)d_in[11];
  float* Y = (float*)d_out;

  float* ws = (float*)d_ws;
  size_t off = 0;
  float* Qh = ws + off; off += (size_t)BH * Ls * Dh;   //  4.19M
  float* Kh = ws + off; off += (size_t)BH * Ls * Dh;
  float* Vh = ws + off; off += (size_t)BH * Ls * Dh;
  float* PQ = ws + off; off += (size_t)R2 * Mm;        // 16.78M (-> q' in place)
  float* PK = ws + off; off += (size_t)R2 * Mm;        // 16.78M (-> k' in place)
  float* G  = ws + off; off += (size_t)BH * NC * Mm * Dh;  // 16.78M (-> prefix in place)
  float* ZS = ws + off; off += (size_t)BH * NC * Mm;
  float* ATT = ws + off; off += (size_t)BH * Ls * Dh;
  float* BM = ws + off; off += 4096;
  float* KS = ws + off; off += 1;

  dim3 blk(256);
  k_proj<<<dim3(DMs / 64, (Bb * Ls) / 64, 3), blk, 0, stream>>>(
      query, key_, value, Wq, Wk, Wv, bq, bk, bv, Qh, Kh, Vh);
  k_phi<<<dim3(Mm / 64, R2 / 64, 2), blk, 0, stream>>>(Qh, Kh, rf, PQ, PK);
  k_max1<<<dim3(4096), blk, 0, stream>>>(PK, BM);
  k_max2<<<dim3(1), blk, 0, stream>>>(BM, KS);
  k_expmap<<<dim3(R2 / 8), blk, 0, stream>>>(PQ, Qh, KS, 1);  // row-max stabilizer
  k_expmap<<<dim3(R2 / 8), blk, 0, stream>>>(PK, Kh, KS, 0);  // global stabilizer
  k_chunksum<<<dim3(Mm / 64, NC, BH), blk, 0, stream>>>(PK, Vh, G, ZS);
  k_prefix<<<dim3(64, BH), blk, 0, stream>>>(G, ZS);
  k_outchunk<<<dim3(NC, BH), blk, 0, stream>>>(PQ, PK, Vh, G, ZS, ATT);
  k_outproj<<<dim3(DMs / 64, (Bb * Ls) / 64), blk, 0, stream>>>(ATT, Wo, bo, Y);
}